// ContrastiveLoss_62105227100871
// MI455X (gfx1250) — compile-verified
//
#include <hip/hip_runtime.h>
#include <math.h>

typedef __attribute__((ext_vector_type(2))) float v2f;
typedef __attribute__((ext_vector_type(8))) float v8f;

#define BB      16
#define DD      64
#define HW      147456          // 384*384
#define NCHUNK  36
#define CHUNK   4096            // HW / NCHUNK
#define PSTRIDE 194             // 64 (c0) + 64 (c1) + 64 (tot) + 2 (cnt)
#define TEMP_INV (1.0f/0.07f)

__device__ __forceinline__ float waveSum(float v) {
#pragma unroll
    for (int off = 16; off > 0; off >>= 1) v += __shfl_down(v, off, 32);
    return v;
}

// ---------------------------------------------------------------------------
// Kernel 1: bandwidth-bound fused pooling.
// Per (b, chunk): stage both label planes in LDS, then each of the 8 waves
// streams 8 feature channels (float4), accumulating
//   s0 = sum f*l0, s1 = sum f*l1, s = sum f
// and writes wave-reduced partials to the workspace. Fully deterministic.
// ---------------------------------------------------------------------------
__global__ void pool_kernel(const float* __restrict__ feat,
                            const float* __restrict__ lab,
                            float* __restrict__ part) {
    __shared__ float4 l0s[CHUNK / 4];
    __shared__ float4 l1s[CHUNK / 4];
    __shared__ float  red[16];

    const int b    = blockIdx.y;
    const int ch   = blockIdx.x;
    const int tid  = threadIdx.x;
    const int lane = tid & 31;
    const int wv   = tid >> 5;
    const size_t hw0 = (size_t)ch * CHUNK;

    const float4* g0 = (const float4*)(lab + ((size_t)b * 2 + 0) * HW + hw0);
    const float4* g1 = (const float4*)(lab + ((size_t)b * 2 + 1) * HW + hw0);
    for (int i = tid; i < CHUNK / 4; i += 256) { l0s[i] = g0[i]; l1s[i] = g1[i]; }
    __syncthreads();

    float* pbase = part + ((size_t)b * NCHUNK + ch) * PSTRIDE;

    // label counts for this chunk
    {
        const float* f0 = (const float*)l0s;
        const float* f1 = (const float*)l1s;
        float c0 = 0.f, c1 = 0.f;
        for (int i = tid; i < CHUNK; i += 256) { c0 += f0[i]; c1 += f1[i]; }
        c0 = waveSum(c0); c1 = waveSum(c1);
        if (lane == 0) { red[wv] = c0; red[8 + wv] = c1; }
        __syncthreads();
        if (tid == 0) {
            float a = 0.f, bsum = 0.f;
            for (int w = 0; w < 8; ++w) { a += red[w]; bsum += red[8 + w]; }
            pbase[192] = a; pbase[193] = bsum;
        }
    }

    // each wave handles one channel per round; 8 rounds cover d = 0..63
    for (int r = 0; r < 8; ++r) {
        const int d = r * 8 + wv;
        const float4* f4 = (const float4*)(feat + ((size_t)b * DD + d) * HW + hw0);
        if (r + 1 < 8) {   // warm next channel row into cache (global_prefetch_b8)
            const char* nxt = (const char*)(feat + ((size_t)b * DD + d + 8) * HW + hw0);
            __builtin_prefetch(nxt + lane * 512, 0, 0);
        }
        float s = 0.f, s0 = 0.f, s1 = 0.f;
        for (int i = lane; i < CHUNK / 4; i += 32) {
            const float4 f  = f4[i];
            const float4 a0 = l0s[i];
            const float4 a1 = l1s[i];
            s  += f.x + f.y + f.z + f.w;
            s0 += f.x * a0.x + f.y * a0.y + f.z * a0.z + f.w * a0.w;
            s1 += f.x * a1.x + f.y * a1.y + f.z * a1.z + f.w * a1.w;
        }
        s = waveSum(s); s0 = waveSum(s0); s1 = waveSum(s1);
        if (lane == 0) {
            pbase[d]       = s0;   // sum_t, c = 0
            pbase[64 + d]  = s1;   // sum_t, c = 1
            pbase[128 + d] = s;    // tot
        }
    }
}

// ---------------------------------------------------------------------------
// Kernel 2: single workgroup. Fold chunk partials, normalize rt/rb, compute
// St = rt*rt^T and Sb = rt*rb^T with V_WMMA_F32_16X16X4_F32 (one 16x16 tile
// per wave, K=64 -> 16 WMMA steps), then the stabilized contrastive loss.
// ---------------------------------------------------------------------------
__global__ void loss_kernel(const float* __restrict__ part,
                            const int* __restrict__ tasks,
                            float* __restrict__ out) {
    __shared__ float sum_t[BB * 2 * DD];   // 2048
    __shared__ float tot  [BB * DD];       // 1024
    __shared__ float cnt  [BB * 2];        // 32
    __shared__ float rt   [32 * DD];       // 2048
    __shared__ float rb   [32 * DD];       // 2048
    __shared__ float St   [32 * 32];
    __shared__ float Sb   [32 * 32];

    const int tid = threadIdx.x;

    // fold NCHUNK partials (fixed order -> deterministic)
    for (int idx = tid; idx < BB * 2 * DD; idx += 256) {
        const int b = idx >> 7, rem = idx & 127;
        float s = 0.f;
        for (int ch = 0; ch < NCHUNK; ++ch)
            s += part[((size_t)b * NCHUNK + ch) * PSTRIDE + rem];
        sum_t[idx] = s;
    }
    for (int idx = tid; idx < BB * DD; idx += 256) {
        const int b = idx >> 6, d = idx & 63;
        float s = 0.f;
        for (int ch = 0; ch < NCHUNK; ++ch)
            s += part[((size_t)b * NCHUNK + ch) * PSTRIDE + 128 + d];
        tot[idx] = s;
    }
    for (int idx = tid; idx < BB * 2; idx += 256) {
        const int b = idx >> 1, c = idx & 1;
        float s = 0.f;
        for (int ch = 0; ch < NCHUNK; ++ch)
            s += part[((size_t)b * NCHUNK + ch) * PSTRIDE + 192 + c];
        cnt[idx] = s;
    }
    __syncthreads();

    // rt / rb rows (row r = 2*b + c), L2-normalized means
    if (tid < 32) {
        const int b = tid >> 1;
        const float ct = fmaxf(cnt[tid], 1.0f);
        const float cb = fmaxf((float)HW - cnt[tid], 1.0f);
        float nt = 0.f, nb = 0.f;
        for (int d = 0; d < DD; ++d) {
            const float st = sum_t[tid * DD + d];
            const float mt = st / ct;
            const float mb = (tot[b * DD + d] - st) / cb;
            nt += mt * mt; nb += mb * mb;
        }
        nt = fmaxf(sqrtf(nt), 1e-12f);
        nb = fmaxf(sqrtf(nb), 1e-12f);
        for (int d = 0; d < DD; ++d) {
            const float st = sum_t[tid * DD + d];
            rt[tid * DD + d] = (st / ct) / nt;
            rb[tid * DD + d] = ((tot[b * DD + d] - st) / cb) / nb;
        }
    }
    __syncthreads();

    // WMMA gram matrices: waves 0..3 -> St tiles, waves 4..7 -> Sb tiles.
    {
        const int wv   = tid >> 5;
        const int lane = tid & 31;
        const int t    = wv & 3;
        const int ti   = t >> 1, tj = t & 1;
        const bool isB = (wv >= 4);
        const float* A  = rt;
        const float* Bm = isB ? rb : rt;
        // A 16x4 f32 fragment: lanes 0-15 hold K=k0,k0+1 ; lanes 16-31 K=k0+2,k0+3.
        // B = (rows of Bm)^T -> identical per-lane indexing with row base tj*16.
        const int rowA = ti * 16 + (lane & 15);
        const int rowB = tj * 16 + (lane & 15);
        const int kh   = (lane >> 4) << 1;     // 0 or 2
        v8f acc = {};
#pragma unroll
        for (int k0 = 0; k0 < DD; k0 += 4) {
            v2f a, bf;
            a.x  = A [rowA * DD + k0 + kh];
            a.y  = A [rowA * DD + k0 + kh + 1];
            bf.x = Bm[rowB * DD + k0 + kh];
            bf.y = Bm[rowB * DD + k0 + kh + 1];
            acc = __builtin_amdgcn_wmma_f32_16x16x4_f32(
                false, a, false, bf, (short)0, acc, false, false);
        }
        // C/D layout: VGPR v, lane L -> D[v + (L>=16 ? 8 : 0)][L & 15]
        float* O = isB ? Sb : St;
        const int mo  = (lane >> 4) * 8;
        const int col = tj * 16 + (lane & 15);
#pragma unroll
        for (int v = 0; v < 8; ++v)
            O[(ti * 16 + mo + v) * 32 + col] = acc[v];
    }
    __syncthreads();

    // stabilized -log(exp(pos)/sum(exp(neg))) per row, then mean
    if (tid < 32) {
        const int i = tid;
        const int myTask = 2 * tasks[i >> 1] + (i & 1);
        float m = -3.4e38f;
        for (int j = 0; j < 32; ++j) {
            const int tk = 2 * tasks[j >> 1] + (j & 1);
            const float v = ((myTask == tk) ? Sb[i * 32 + j] : St[i * 32 + j]) * TEMP_INV;
            m = fmaxf(m, v);
        }
        float se = 0.f; int pidx = -1;
        for (int j = 0; j < 32; ++j) {
            const int tk = 2 * tasks[j >> 1] + (j & 1);
            const bool same = (myTask == tk);
            const float v = (same ? Sb[i * 32 + j] : St[i * 32 + j]) * TEMP_INV;
            se += expf(v - m);
            if (same && (j != i) && (pidx < 0)) pidx = j;
        }
        if (pidx < 0) pidx = 0;
        const float pos = St[i * 32 + pidx] * TEMP_INV;
        float li = logf(se) - (pos - m);
        li = waveSum(li);
        if (i == 0) out[0] = li * (1.0f / 32.0f);
    }
}

extern "C" void kernel_launch(void* const* d_in, const int* in_sizes, int n_in,
                              void* d_out, int out_size, void* d_ws, size_t ws_size,
                              hipStream_t stream) {
    const float* feat  = (const float*)d_in[0];   // [16,64,384,384] f32
    const float* lab   = (const float*)d_in[1];   // [16,2,384,384]  f32
    const int*   tasks = (const int*)  d_in[2];   // [16] i32
    float* part = (float*)d_ws;                   // 16*36*194 floats (~447 KB)

    dim3 grid(NCHUNK, BB);
    pool_kernel<<<grid, 256, 0, stream>>>(feat, lab, part);
    loss_kernel<<<1, 256, 0, stream>>>(part, tasks, (float*)d_out);
}